// RadarSparseProcessor_91104846283338
// MI455X (gfx1250) — compile-verified
//
#include <hip/hip_runtime.h>

typedef __attribute__((ext_vector_type(2))) float v2f;
typedef __attribute__((ext_vector_type(8))) float v8f;

// out[v, o] = (1/max(n_v,1)) * sum_c W[o,c] * (sum_p feat[v,p,c])
// V voxels, P=4 points, C_IN=4, C_OUT=32.
// One wave32 processes 16 voxels with two V_WMMA_F32_16X16X4_F32 ops.
__global__ void __launch_bounds__(256)
radar_pointnet_wmma(const float* __restrict__ vf,   // [V,4,4] fp32
                    const float* __restrict__ W,    // [32,4] fp32 row-major
                    const int*   __restrict__ npts, // per-voxel point count
                    float*       __restrict__ out,  // [V,32] fp32
                    int V, int numTiles)
{
    const int lane      = threadIdx.x & 31;
    const int waveInBlk = threadIdx.x >> 5;

    // Wave-uniform tile index, made *provably* scalar so all tile-level branches
    // compile to s_cmp/s_cbranch (no EXEC manipulation around the WMMAs/stores).
    const int tile = __builtin_amdgcn_readfirstlane(
        blockIdx.x * (blockDim.x >> 5) + waveInBlk);
    if (tile >= numTiles) return;                 // scalar branch, EXEC untouched

    const int  v0 = tile << 4;                    // first voxel of this 16-voxel tile (SGPR)
    const int  lm = lane & 15;
    const bool hi = lane >= 16;

    // ---------------- B operands: two 16-col tiles of W^T (4x16, K-major) ----------------
    // VGPR0 holds K=0 (lanes 0-15) / K=2 (lanes 16-31); VGPR1 holds K=1 / K=3.
    const float4 w0 = ((const float4*)W)[lm];       // W rows 0..15  -> out cols 0..15
    const float4 w1 = ((const float4*)W)[16 + lm];  // W rows 16..31 -> out cols 16..31
    v2f b0, b1;
    b0.x = hi ? w0.z : w0.x;  b0.y = hi ? w0.w : w0.y;
    b1.x = hi ? w1.z : w1.x;  b1.y = hi ? w1.w : w1.y;

    // ---------------- A operand: point-sum, normalized ----------------
    int v = v0 + lm;
    if (v >= V) v = V - 1;                        // clamp loads (only ragged tile)
    const float4* f = (const float4*)(vf + (size_t)v * 16);  // 4 points x 4 ch = 64 B
    const float4 p0 = f[0], p1 = f[1], p2 = f[2], p3 = f[3];
    const float sx = (p0.x + p1.x) + (p2.x + p3.x);
    const float sy = (p0.y + p1.y) + (p2.y + p3.y);
    const float sz = (p0.z + p1.z) + (p2.z + p3.z);
    const float sw = (p0.w + p1.w) + (p2.w + p3.w);

    int n = npts[v];
    if (n < 1) n = 1;
    const float inv = 1.0f / (float)n;            // matches reference fp32 division

    // A 16x4: VGPR0 = K0 (lanes 0-15) / K2 (lanes 16-31); VGPR1 = K1 / K3.
    v2f a;
    a.x = (hi ? sz : sx) * inv;
    a.y = (hi ? sw : sy) * inv;

    // ---------------- D = A x B (+0) : two 16x16 output tiles ----------------
    v8f zero = {0.f, 0.f, 0.f, 0.f, 0.f, 0.f, 0.f, 0.f};
    v8f d0 = __builtin_amdgcn_wmma_f32_16x16x4_f32(
        /*neg_a=*/false, a, /*neg_b=*/false, b0, /*c_mod=*/(short)0, zero,
        /*reuse_a=*/false, /*reuse_b=*/false);
    v8f d1 = __builtin_amdgcn_wmma_f32_16x16x4_f32(
        /*neg_a=*/false, a, /*neg_b=*/false, b1, /*c_mod=*/(short)0, zero,
        /*reuse_a=*/false, /*reuse_b=*/false);

    // ---------------- Store D (streaming / non-temporal) ----------------
    // D VGPR r: lanes 0-15 -> row v0+r, col lm ; lanes 16-31 -> row v0+r+8, col lm.
    const int rbase = v0 + (hi ? 8 : 0);
    float* const p = out + (size_t)rbase * 32 + lm;   // one base addr, imm offsets below

    if (v0 + 16 <= V) {                           // scalar condition (v0 is SGPR)
        // Fast path (always taken when V % 16 == 0): one clause of 16 NT stores.
#pragma unroll
        for (int r = 0; r < 8; ++r) {
            __builtin_nontemporal_store(d0[r], p + r * 32);        // cols  0..15
            __builtin_nontemporal_store(d1[r], p + r * 32 + 16);   // cols 16..31
        }
    } else {
        // Ragged tail tile: per-row guard (cold).
#pragma unroll
        for (int r = 0; r < 8; ++r) {
            if (rbase + r < V) {
                __builtin_nontemporal_store(d0[r], p + r * 32);
                __builtin_nontemporal_store(d1[r], p + r * 32 + 16);
            }
        }
    }
}

extern "C" void kernel_launch(void* const* d_in, const int* in_sizes, int n_in,
                              void* d_out, int out_size, void* d_ws, size_t ws_size,
                              hipStream_t stream)
{
    const float* vf = (const float*)d_in[0];   // voxel_features [V,4,4]
    const float* W  = (const float*)d_in[1];   // W [32,4]
    const int*   np = (const int*)d_in[2];     // voxel_num_points [V]
    float*       out = (float*)d_out;          // [V,32]

    const int V        = in_sizes[0] / 16;     // V * P(4) * C_IN(4)
    const int numTiles = (V + 15) / 16;        // 16 voxels per wave
    const int wavesPerBlock = 8;               // 256 threads
    const int blocks = (numTiles + wavesPerBlock - 1) / wavesPerBlock;

    radar_pointnet_wmma<<<blocks, 256, 0, stream>>>(vf, W, np, out, V, numTiles);
}